// QuantumFeatureExtractor_65481071396886
// MI455X (gfx1250) — compile-verified
//
#include <hip/hip_runtime.h>
#include <hip/hip_bf16.h>

// QuantumFeatureExtractor: the reference circuit (20 independent RX gates on
// |0...0>, then per-wire <Z>) stays a product state, so the marginal
// expectation factorizes exactly:
//     out[b, i] = cos(x[b, i] + theta[i])
// 320-element elementwise cosine instead of a 16 x 2^20 complex64 simulation
// (~5 GB HBM traffic ~= 220 us at 23.3 TB/s). This kernel is launch-latency
// bound (~2.6 KB total traffic), so the optimization goal is a minimal
// instruction stream:
//   - block shaped (20, 16): wire index = threadIdx.x (no div/mod magic)
//   - v_cos_f32 via __builtin_amdgcn_cosf (input in revolutions): hardware
//     cosine is accurate over |rev| << 256; our args are |a| <~ 8 rad, and
//     the 1/2pi pre-scale rounding (<= ~6e-7 rad) is far below the fp32
//     error the reference itself accumulates over 2^19-term marginal sums.
//   - no LDS/barrier: 320 lanes reading 20 theta floats hit one cacheline.

#define QFE_BATCH 16
#define QFE_NQ    20
#define QFE_TOTAL (QFE_BATCH * QFE_NQ)   // 320 threads = 10 wave32s, one WGP

__global__ __launch_bounds__(QFE_TOTAL)
void QuantumFeatureExtractor_65481071396886_kernel(
    const float* __restrict__ x,      // [BATCH, NQ] row-major
    const float* __restrict__ theta,  // [NQ]
    float* __restrict__ out)          // [BATCH, NQ] row-major
{
    const int i   = threadIdx.x;            // wire  0..19 (from packed VGPR0)
    const int b   = threadIdx.y;            // batch 0..15
    const int idx = b * QFE_NQ + i;

    const float a = x[idx] + theta[i];
    // v_cos_f32 takes its argument in revolutions (1.0 == 2*pi radians).
    const float r = a * 0.15915494309189535f;   // 1/(2*pi)
    out[idx] = __builtin_amdgcn_cosf(r);
}

extern "C" void kernel_launch(void* const* d_in, const int* in_sizes, int n_in,
                              void* d_out, int out_size, void* d_ws, size_t ws_size,
                              hipStream_t stream) {
    (void)in_sizes; (void)n_in; (void)d_ws; (void)ws_size; (void)out_size;
    const float* x     = (const float*)d_in[0];   // [16, 20] float32
    const float* theta = (const float*)d_in[1];   // [20]     float32
    float*       out   = (float*)d_out;           // [16, 20] float32

    QuantumFeatureExtractor_65481071396886_kernel<<<1, dim3(QFE_NQ, QFE_BATCH), 0, stream>>>(
        x, theta, out);
}